// FCHLCuda_87608742903887
// MI455X (gfx1250) — compile-verified
//
#include <hip/hip_runtime.h>
#include <math.h>

// ---------------------------------------------------------------------------
// FCHL19-style representation for MI455X (gfx1250, wave32).
// One wave per atom (b,i). rep2 species-contraction done with
// V_WMMA_F32_16X16X4_F32 (the reference's one-hot einsum IS this matmul);
// rep3 pair-channel scatter via ds_add_f32 LDS atomics.
// Fragment construction is fully branchless: LDS tails are zero-filled so
// out-of-range neighbors contribute exactly zero, and per-lane masks are
// folded with cndmask instead of EXEC-manipulating branches.
// ---------------------------------------------------------------------------

#define NSPEC   4
#define NRS2    24
#define NRS3    20
#define NPAIRS  10          // NSPEC*(NSPEC+1)/2
#define RCUT    6.0f
#define ETA2    0.32f
#define ETA3    2.7f
#define PI_F    3.14159265358979323846f
#define W3      12.4225788f          // sqrt(2.7/pi)*13.4
#define INV_SQRT_2PI 0.3989422804014327f
#define MAXA    64                   // LDS sizing (A = 48 in reference)
#define REP2    (NSPEC * NRS2)       // 96
#define REP3    (NPAIRS * NRS3 * 2)  // 400
#define REPTOT  (REP2 + REP3)        // 496

typedef __attribute__((ext_vector_type(2))) float v2f;
typedef __attribute__((ext_vector_type(8))) float v8f;

__global__ __launch_bounds__(32)
void fchl_rep_kernel(const float* __restrict__ X,
                     const int*   __restrict__ Z,
                     const int*   __restrict__ atom_counts,
                     float*       __restrict__ out,
                     int B, int A)
{
    const int gid  = blockIdx.x;          // b*A + i
    const int b    = gid / A;
    const int i    = gid - b * A;
    const int lane = threadIdx.x;

    __shared__ float sD[MAXA * 3];   // xi - xj
    __shared__ float sR[MAXA];       // rs value: r if within else 1.0
    __shared__ float sFC[MAXA];      // cutoff function (0 if !within)
    __shared__ float sPref[MAXA];    // 2-body prefactor (0 if !within)
    __shared__ float sMu[MAXA];      // log-normal mean
    __shared__ float sI2S[MAXA];     // 1/(2*sigma^2)  (0 if !within)
    __shared__ int   sZ[MAXA];
    __shared__ int   sWn[MAXA];      // within flag
    __shared__ float sAcc[REP3];     // rep3 accumulator [pair][shell][cos/sin]

    const int  nA = atom_counts[b];
    const bool vi = (i < nA);

    const float xi0 = X[(size_t)(b * A + i) * 3 + 0];
    const float xi1 = X[(size_t)(b * A + i) * 3 + 1];
    const float xi2 = X[(size_t)(b * A + i) * 3 + 2];

    // ---------------- Phase 0: per-neighbor scalars into LDS ----------------
    // Fill the full MAXA extent so later unguarded LDS reads (j >= A) are
    // well-defined and contribute exactly zero.
    for (int j = lane; j < MAXA; j += 32) {
        float d0 = 0.0f, d1 = 0.0f, d2 = 0.0f;
        bool  within = false;
        float r = 1.0f, fc = 0.0f, pref = 0.0f, mu = 0.0f, i2s = 0.0f;
        int   zj = -1;

        if (j < A) {
            d0 = xi0 - X[(size_t)(b * A + j) * 3 + 0];
            d1 = xi1 - X[(size_t)(b * A + j) * 3 + 1];
            d2 = xi2 - X[(size_t)(b * A + j) * 3 + 2];
            const float r2 = d0 * d0 + d1 * d1 + d2 * d2;
            const bool  pm = vi && (j < nA) && (j != i);
            const float rr = sqrtf(pm ? r2 : 1.0f);
            within = pm && (rr < RCUT);
            if (within) {
                r  = rr;
                fc = 0.5f * (cosf(PI_F * rr / RCUT) + 1.0f);
                const float s2 = log1pf(ETA2 / (rr * rr));
                mu   = logf(rr) - 0.5f * s2;
                pref = fc * powf(rr, -1.8f) * (1.0f / sqrtf(s2)) * INV_SQRT_2PI;
                i2s  = 0.5f / s2;
            }
            zj = Z[b * A + j];
        }
        sD[j * 3 + 0] = d0;
        sD[j * 3 + 1] = d1;
        sD[j * 3 + 2] = d2;
        sR[j]    = r;
        sFC[j]   = fc;
        sPref[j] = pref;
        sMu[j]   = mu;
        sI2S[j]  = i2s;
        sZ[j]    = zj;
        sWn[j]   = within ? 1 : 0;
    }
    __syncthreads();

    const size_t outBase = (size_t)gid * REPTOT;
    const int half = lane >> 4;
    const int m    = lane & 15;

    // ---------------- Phase 1: rep2 = onehot(Z)^T @ radial2 via WMMA --------
    // D(16x16) += A(16x4) x B(4x16): M=species (rows 0..3 used),
    // N=radial shell (two 16-wide tiles covering NRS2=24), K=neighbor j.
    // A frag (ISA 7.12.2, 32-bit 16x4): lane=M; V0 holds K=(half?2:0), V1 K+1.
    // B frag (4x16): V0 holds row K=(half?2:0) at col N=lane&15, V1 row K+1.
    const int   nChunks = (A + 3) >> 2;            // LDS tail handles overrun
    const float mMask   = (m < NSPEC) ? 1.0f : 0.0f;  // species-row mask
    const int   jbase   = half ? 2 : 0;

    for (int t = 0; t < 2; ++t) {
        const int   kidx    = t * 16 + m;
        const bool  kval    = (kidx < NRS2);
        const float rs2k    = 0.25f * (float)(kidx + 1);
        const float lnrs2k  = kval ? logf(rs2k) : 0.0f;
        const float invrs2k = kval ? (1.0f / rs2k) : 0.0f;  // zeroes padded shells

        v8f c = {};
        #pragma unroll 4
        for (int jc = 0; jc < nChunks; ++jc) {
            const int j0 = jc * 4 + jbase;
            const int j1 = j0 + 1;

            // Branchless: sZ tail = -1 (never matches), sPref tail = 0.
            v2f a, bb;
            a.x = (sZ[j0] == m) ? mMask : 0.0f;
            a.y = (sZ[j1] == m) ? mMask : 0.0f;

            const float u0 = lnrs2k - sMu[j0];
            const float u1 = lnrs2k - sMu[j1];
            bb.x = sPref[j0] * invrs2k * expf(-u0 * u0 * sI2S[j0]);
            bb.y = sPref[j1] * invrs2k * expf(-u1 * u1 * sI2S[j1]);

            // 8 args: (neg_a, A, neg_b, B, c_mod, C, reuse_a, reuse_b)
            c = __builtin_amdgcn_wmma_f32_16x16x4_f32(
                    false, a, false, bb, (short)0, c, false, false);
        }
        // D layout: VGPR v, lanes 0..15 -> row M=v, col N=lane&15.
        if (half == 0 && kval) {
            out[outBase + 0 * NRS2 + kidx] = c[0];
            out[outBase + 1 * NRS2 + kidx] = c[1];
            out[outBase + 2 * NRS2 + kidx] = c[2];
            out[outBase + 3 * NRS2 + kidx] = c[3];
        }
    }

    // ---------------- Phase 2: rep3 (triples j<k, lanes over k) -------------
    for (int idx = lane; idx < REP3; idx += 32) sAcc[idx] = 0.0f;
    __syncthreads();

    if (vi) {
        for (int j = 0; j < A - 1; ++j) {
            if (!sWn[j]) continue;
            const float rij = sR[j];
            const float fcj = sFC[j];
            const float dj0 = sD[j * 3 + 0];
            const float dj1 = sD[j * 3 + 1];
            const float dj2 = sD[j * 3 + 2];
            const int   zj  = sZ[j];
            const float rij2 = rij * rij;

            for (int k = j + 1 + lane; k < A; k += 32) {
                if (!sWn[k]) continue;
                const float rik = sR[k];
                const float dk0 = sD[k * 3 + 0];
                const float dk1 = sD[k * 3 + 1];
                const float dk2 = sD[k * 3 + 2];

                // xk - xj = (xi - xj) - (xi - xk)
                const float e0 = dj0 - dk0, e1 = dj1 - dk1, e2 = dj2 - dk2;
                float rjk2 = e0 * e0 + e1 * e1 + e2 * e2;
                float rjk  = sqrtf(rjk2);
                if (rjk >= RCUT) { rjk = 1.0f; rjk2 = 1.0f; }  // reference's rs(j,k) quirk

                float ci = (dj0 * dk0 + dj1 * dk1 + dj2 * dk2) / (rij * rik);
                ci = fminf(1.0f, fmaxf(-1.0f, ci));
                const float rik2 = rik * rik;
                const float cj = (rij2 + rjk2 - rik2) / (2.0f * rij * rjk);
                const float ck = (rik2 + rjk2 - rij2) / (2.0f * rik * rjk);

                const float ksi = W3 * (1.0f + 3.0f * ci * cj * ck)
                                * powf(rij * rik * rjk, -0.57f);
                const float w   = ksi * fcj * sFC[k];
                const float wc  = w * ci;
                const float wsn = w * sqrtf(fminf(1.0f, fmaxf(0.0f, 1.0f - ci * ci)));
                const float d   = 0.5f * (rij + rik);

                const int zk   = sZ[k];
                const int pmn  = (zj < zk) ? zj : zk;
                const int pmx  = (zj < zk) ? zk : zj;
                const int pidx = (pmn * (2 * NSPEC - 1 - pmn)) / 2 + pmx;
                float* acc = &sAcc[pidx * NRS3 * 2];

                #pragma unroll
                for (int l = 0; l < NRS3; ++l) {
                    const float dr  = d - 0.3f * (float)(l + 1);
                    const float rad = expf(-ETA3 * dr * dr);
                    atomicAdd(&acc[2 * l + 0], wc  * rad);   // ds_add_f32
                    atomicAdd(&acc[2 * l + 1], wsn * rad);
                }
            }
        }
    }
    __syncthreads();

    for (int idx = lane; idx < REP3; idx += 32)
        out[outBase + REP2 + idx] = sAcc[idx];
}

extern "C" void kernel_launch(void* const* d_in, const int* in_sizes, int n_in,
                              void* d_out, int out_size, void* d_ws, size_t ws_size,
                              hipStream_t stream) {
    // setup_inputs order: X, Z, atomIDs, molIDs, atom_counts
    const float* X           = (const float*)d_in[0];
    const int*   Z           = (const int*)  d_in[1];
    const int*   atom_counts = (const int*)  d_in[4];
    float*       out         = (float*)d_out;

    const int B = in_sizes[4];            // atom_counts has B entries
    const int A = in_sizes[1] / B;        // Z is B*A

    dim3 grid(B * A);
    dim3 block(32);                       // one full wave32 -> EXEC all-ones for WMMA
    fchl_rep_kernel<<<grid, block, 0, stream>>>(X, Z, atom_counts, out, B, A);
}